// MattingLaplacian_51728586113164
// MI455X (gfx1250) — compile-verified
//
#include <hip/hip_runtime.h>
#include <stdint.h>

typedef float v2f __attribute__((ext_vector_type(2)));
typedef float v8f __attribute__((ext_vector_type(8)));
typedef int gv4i __attribute__((vector_size(16)));   // pointee type for async b128
typedef int gv2i __attribute__((vector_size(8)));    // pointee type for async b64

#define AS1 __attribute__((address_space(1)))
#define AS3 __attribute__((address_space(3)))

// ---- feature detection (compile-safe hedges) -------------------------------
#if __has_builtin(__builtin_amdgcn_global_load_async_to_lds_b128) && \
    __has_builtin(__builtin_amdgcn_global_load_async_to_lds_b64)
  #define ASYNC_MODE 2
#elif __has_builtin(__builtin_amdgcn_global_load_async_to_lds_b32)
  #define ASYNC_MODE 1
#else
  #define ASYNC_MODE 0
#endif

#if __has_builtin(__builtin_amdgcn_wmma_f32_16x16x4_f32)
  #define HAVE_WMMA_F32_16X16X4 1
#else
  #define HAVE_WMMA_F32_16X16X4 0
#endif

__device__ __forceinline__ void wait_async_lds() {
#if ASYNC_MODE
  #if __has_builtin(__builtin_amdgcn_s_wait_asynccnt)
    __builtin_amdgcn_s_wait_asynccnt(0);
  #else
    asm volatile("s_wait_asynccnt 0" ::: "memory");
  #endif
#endif
}

namespace {
constexpr int   WIMG   = 512;
constexpr int   HIMG   = 512;
constexpr int   HWIMG  = WIMG * HIMG;
constexpr int   NWIN   = WIMG - 2;                 // 510 windows per dim
constexpr int   TILE   = 16;                       // 16x16 windows per block
constexpr int   NT     = (NWIN + TILE - 1) / TILE; // 32 tiles per dim
constexpr int   PITCH  = 20;                       // LDS row pitch (floats), 80B => 16B aligned
constexpr float LAMBDA = 1e-7f / 9.0f;             // EPS / WIN_SIZE
}

// One block: 16x16 windows of one image. Stage 18x18x3 pixels in LDS,
// per-thread 3x3 symmetric solve, WMMA ones-matmul block reduction.
__global__ __launch_bounds__(256) void matting_laplacian_kernel(
    const float* __restrict__ img, float* __restrict__ out, float inv_batch)
{
  __shared__ float sT[3 * 18 * PITCH];   // [ch][row][col(padded)]
  __shared__ float sRed[256];

  const int t  = threadIdx.x;
  const int x0 = blockIdx.x * TILE;
  const int y0 = blockIdx.y * TILE;
  const int ncols = (WIMG - x0 < 18) ? (WIMG - x0) : 18;  // 18, or 16 at the edge
  const int nrows = (HIMG - y0 < 18) ? (HIMG - y0) : 18;
  const float* __restrict__ base = img + (size_t)blockIdx.z * 3 * HWIMG;

  // ---- stage tile into LDS ------------------------------------------------
#if ASYNC_MODE == 2
  if (t < 54) {                       // one lane per (channel, row)
    const int ch = t / 18, row = t % 18;
    if (row < nrows) {
      const float* g = base + ch * HWIMG + (y0 + row) * WIMG + x0;  // 64B aligned
      float*       l = &sT[(ch * 18 + row) * PITCH];                // 80B pitch => 16B aligned
      __builtin_amdgcn_global_load_async_to_lds_b128((AS1 gv4i*)(uintptr_t)(g),      (AS3 gv4i*)(l),      0, 0);
      __builtin_amdgcn_global_load_async_to_lds_b128((AS1 gv4i*)(uintptr_t)(g + 4),  (AS3 gv4i*)(l + 4),  0, 0);
      __builtin_amdgcn_global_load_async_to_lds_b128((AS1 gv4i*)(uintptr_t)(g + 8),  (AS3 gv4i*)(l + 8),  0, 0);
      __builtin_amdgcn_global_load_async_to_lds_b128((AS1 gv4i*)(uintptr_t)(g + 12), (AS3 gv4i*)(l + 12), 0, 0);
      if (ncols == 18)
        __builtin_amdgcn_global_load_async_to_lds_b64((AS1 gv2i*)(uintptr_t)(g + 16), (AS3 gv2i*)(l + 16), 0, 0);
    }
  }
#elif ASYNC_MODE == 1
  #pragma unroll
  for (int i = 0; i < 4; ++i) {
    const int e = t + 256 * i;
    if (e < 3 * 18 * 18) {
      const int ch = e / 324, rem = e % 324, row = rem / 18, col = rem % 18;
      if (row < nrows && col < ncols) {
        const float* g = base + ch * HWIMG + (y0 + row) * WIMG + (x0 + col);
        float*       l = &sT[(ch * 18 + row) * PITCH + col];
        __builtin_amdgcn_global_load_async_to_lds_b32((AS1 int*)(uintptr_t)g, (AS3 int*)l, 0, 0);
      }
    }
  }
#else
  #pragma unroll
  for (int i = 0; i < 4; ++i) {
    const int e = t + 256 * i;
    if (e < 3 * 18 * 18) {
      const int ch = e / 324, rem = e % 324, row = rem / 18, col = rem % 18;
      if (row < nrows && col < ncols)
        sT[(ch * 18 + row) * PITCH + col] =
            base[ch * HWIMG + (y0 + row) * WIMG + (x0 + col)];
    }
  }
#endif
  wait_async_lds();
  __syncthreads();

  // ---- per-window: loss_w = 9*lambda * tr((Sigma+lambda I)^-1 Sigma) ------
  const int lx = t & 15, ly = t >> 4;
  float loss = 0.0f;
  if ((x0 + lx) < NWIN && (y0 + ly) < NWIN) {
    float s0 = 0.f, s1 = 0.f, s2 = 0.f;
    float m00 = 0.f, m01 = 0.f, m02 = 0.f, m11 = 0.f, m12 = 0.f, m22 = 0.f;
    #pragma unroll
    for (int dy = 0; dy < 3; ++dy) {
      const int r = ly + dy;
      #pragma unroll
      for (int dx = 0; dx < 3; ++dx) {
        const int c = lx + dx;
        const float p0 = sT[(0 * 18 + r) * PITCH + c];
        const float p1 = sT[(1 * 18 + r) * PITCH + c];
        const float p2 = sT[(2 * 18 + r) * PITCH + c];
        s0 += p0; s1 += p1; s2 += p2;
        m00 = fmaf(p0, p0, m00); m01 = fmaf(p0, p1, m01); m02 = fmaf(p0, p2, m02);
        m11 = fmaf(p1, p1, m11); m12 = fmaf(p1, p2, m12); m22 = fmaf(p2, p2, m22);
      }
    }
    const float k   = 1.0f / 9.0f;
    const float mu0 = s0 * k, mu1 = s1 * k, mu2 = s2 * k;
    const float c00 = fmaf(m00, k, -mu0 * mu0);
    const float c01 = fmaf(m01, k, -mu0 * mu1);
    const float c02 = fmaf(m02, k, -mu0 * mu2);
    const float c11 = fmaf(m11, k, -mu1 * mu1);
    const float c12 = fmaf(m12, k, -mu1 * mu2);
    const float c22 = fmaf(m22, k, -mu2 * mu2);
    const float a00 = c00 + LAMBDA, a11 = c11 + LAMBDA, a22 = c22 + LAMBDA;
    // adjugate of symmetric A
    const float d00 = a11 * a22 - c12 * c12;
    const float d01 = c02 * c12 - c01 * a22;
    const float d02 = c01 * c12 - c02 * a11;
    const float d11 = a00 * a22 - c02 * c02;
    const float d12 = c01 * c02 - a00 * c12;
    const float d22 = a00 * a11 - c01 * c01;
    const float det = a00 * d00 + c01 * d01 + c02 * d02;
    const float tr  = d00 * c00 + d11 * c11 + d22 * c22
                    + 2.0f * (d01 * c01 + d02 * c02 + d12 * c12);
    loss = (9.0f * LAMBDA) * (tr / det);
  }
  sRed[t] = loss;
  __syncthreads();

  // ---- block reduction: 256 -> 64 partials -> one WMMA (ones matmul) ------
  if (t < 32) {   // wave 0 only; EXEC all ones as WMMA requires
    const float pA = sRed[t]      + sRed[t + 64] + sRed[t + 128] + sRed[t + 192];
    const float pB = sRed[t + 32] + sRed[t + 96] + sRed[t + 160] + sRed[t + 224];
#if HAVE_WMMA_F32_16X16X4
    // A(16x4) holds the 64 partials; B(4x16)=ones (layout-invariant);
    // D[m,n] = row-sum m.  Lanes 0-15 hold rows 0-7, lanes 16-31 rows 8-15.
    v2f a;  a.x = pA;  a.y = pB;
    v2f bOnes;  bOnes.x = 1.0f;  bOnes.y = 1.0f;
    v8f cz = {};
    v8f d = __builtin_amdgcn_wmma_f32_16x16x4_f32(false, a, false, bOnes,
                                                  (short)0, cz, false, false);
    float rs = 0.f;
    #pragma unroll
    for (int i = 0; i < 8; ++i) rs += d[i];
    const float tot = rs + __shfl_xor(rs, 16, 32);   // rows 0-7 + rows 8-15
    if (t == 0) atomicAdd(out, tot * inv_batch);
#else
    float v = pA + pB;
    #pragma unroll
    for (int off = 16; off > 0; off >>= 1) v += __shfl_down(v, off, 32);
    if (t == 0) atomicAdd(out, v * inv_batch);
#endif
  }
}

__global__ void zero_kernel(float* out, int n) {
  const int i = blockIdx.x * blockDim.x + threadIdx.x;
  if (i < n) out[i] = 0.0f;
}

extern "C" void kernel_launch(void* const* d_in, const int* in_sizes, int n_in,
                              void* d_out, int out_size, void* d_ws, size_t ws_size,
                              hipStream_t stream) {
  (void)n_in; (void)d_ws; (void)ws_size;
  const float* img = (const float*)d_in[0];
  float* out = (float*)d_out;
  const int B = in_sizes[0] / (3 * WIMG * HIMG);   // 4
  zero_kernel<<<1, 64, 0, stream>>>(out, out_size);
  dim3 grid(NT, NT, B);
  matting_laplacian_kernel<<<grid, dim3(256), 0, stream>>>(img, out, 1.0f / (float)B);
}